// Router_89172111000181
// MI455X (gfx1250) — compile-verified
//
#include <hip/hip_runtime.h>

// ---------------------------------------------------------------------------
// MoE router for MI455X (gfx1250, wave32).
// Outputs (concatenated in d_out): logits (T*E f32), dispatch (T*E f32),
// expert_indices (E*T i32, sorted ascending, padded with -1).
// ---------------------------------------------------------------------------

typedef __attribute__((ext_vector_type(2))) float v2f;
typedef __attribute__((ext_vector_type(4))) float v4f;
typedef __attribute__((ext_vector_type(8))) float v8f;

#define D_DIM 4096
#define E_DIM 16
#define KC 512              // K-chunk of W staged in LDS (16*512*4 = 32 KB)
#define TOK_PER_BLOCK 128   // 8 waves * 16 tokens
#define THREADS 256

__global__ __launch_bounds__(THREADS)
void router_gemm_top2(const float* __restrict__ x,
                      const float* __restrict__ W,
                      const float* __restrict__ bias_vec,
                      float* __restrict__ logits_out,
                      float* __restrict__ dispatch_out,
                      int T)
{
    __shared__ float lds_w[E_DIM * KC];                 // W chunk, row-major per expert
    __shared__ float lds_logits[TOK_PER_BLOCK * 17];    // stride 17: no bank conflicts

    const int tid  = threadIdx.x;
    const int wave = tid >> 5;
    const int lane = tid & 31;
    const int half = lane >> 4;      // 0: lanes 0-15, 1: lanes 16-31
    const int l16  = lane & 15;
    const int tileBase = blockIdx.x * TOK_PER_BLOCK + wave * 16;

    v8f acc = {0.f, 0.f, 0.f, 0.f, 0.f, 0.f, 0.f, 0.f};

    for (int kc = 0; kc < D_DIM; kc += KC) {
        // Cooperative, coalesced b128 load of W[:, kc:kc+KC] into LDS.
        for (int q = tid; q < (E_DIM * KC) / 4; q += THREADS) {
            const int f  = q * 4;
            const int e  = f / KC;
            const int kk = f - e * KC;
            *(v4f*)&lds_w[e * KC + kk] =
                *(const v4f*)&W[(size_t)e * D_DIM + kc + kk];
        }
        __syncthreads();

        // A (x tile): lane-half picks the K pair; B (W) mirrors the layout.
        // One float4 per lane for A and B covers 8 K-columns via 2 WMMAs:
        //   wmma(.xy) -> K {k,k+1,k+4,k+5}, wmma(.zw) -> K {k+2,k+3,k+6,k+7}.
        const float* xrow = x + (size_t)(tileBase + l16) * D_DIM + kc + 4 * half;
        const float* wrow = lds_w + l16 * KC + 4 * half;

        #pragma unroll 4
        for (int k = 0; k < KC; k += 8) {
            v4f av = *(const v4f*)(xrow + k);
            v4f wv = *(const v4f*)(wrow + k);
            v2f a0 = {av.x, av.y};
            v2f a1 = {av.z, av.w};
            v2f b0 = {wv.x, wv.y};
            v2f b1 = {wv.z, wv.w};
            acc = __builtin_amdgcn_wmma_f32_16x16x4_f32(
                false, a0, false, b0, (short)0, acc, false, false);
            acc = __builtin_amdgcn_wmma_f32_16x16x4_f32(
                false, a1, false, b1, (short)0, acc, false, false);
        }
        __syncthreads();
    }

    // Epilogue: C layout -> lane holds expert column e = l16, rows m = g + 8*half.
    const float bias = bias_vec[l16];
    #pragma unroll
    for (int g = 0; g < 8; ++g) {
        const int m = g + 8 * half;
        lds_logits[(wave * 16 + m) * 17 + l16] = acc[g] + bias;
    }
    __syncthreads();

    // Scalar per-token phase: 1 thread = 1 token (E=16 is tiny).
    if (tid < TOK_PER_BLOCK) {
        const int token = blockIdx.x * TOK_PER_BLOCK + tid;
        float l[E_DIM];
        #pragma unroll
        for (int e = 0; e < E_DIM; ++e) l[e] = lds_logits[tid * 17 + e];

        // Write router logits (coalesced float4 stores).
        #pragma unroll
        for (int e4 = 0; e4 < E_DIM; e4 += 4) {
            v4f v = {l[e4], l[e4 + 1], l[e4 + 2], l[e4 + 3]};
            *(v4f*)&logits_out[(size_t)token * E_DIM + e4] = v;
        }

        // Top-2 (first occurrence wins on ties, matching jax top_k).
        float m1 = -3.402823466e38f; int i1 = 0;
        #pragma unroll
        for (int e = 0; e < E_DIM; ++e) { if (l[e] > m1) { m1 = l[e]; i1 = e; } }
        float m2 = -3.402823466e38f; int i2 = 0;
        #pragma unroll
        for (int e = 0; e < E_DIM; ++e) {
            if (e != i1 && l[e] > m2) { m2 = l[e]; i2 = e; }
        }

        // Renormalized top-2 softmax weights depend only on the logit gap.
        const float t  = __expf(m2 - m1);
        const float w1 = 1.0f / (1.0f + t);
        const float w2 = 1.0f - w1;

        float d[E_DIM];
        #pragma unroll
        for (int e = 0; e < E_DIM; ++e) d[e] = 0.0f;
        d[i1] = w1;
        d[i2] = w2;
        #pragma unroll
        for (int e4 = 0; e4 < E_DIM; e4 += 4) {
            v4f v = {d[e4], d[e4 + 1], d[e4 + 2], d[e4 + 3]};
            *(v4f*)&dispatch_out[(size_t)token * E_DIM + e4] = v;
        }
    }
}

// ---------------------------------------------------------------------------
// Per-expert stable compaction: out[e][pos] = ascending token ids with
// dispatch[t][e] > 0, padded with -1.  One block per expert; wave32 ballot +
// popcount prefix + cross-wave LDS scan.
// ---------------------------------------------------------------------------
__global__ __launch_bounds__(1024)
void expert_index_scan(const float* __restrict__ dispatch,
                       int* __restrict__ out_idx,
                       int T)
{
    __shared__ int waveCnt[32];
    const int e    = blockIdx.x;
    const int tid  = threadIdx.x;
    const int wave = tid >> 5;
    const int lane = tid & 31;

    int base = 0;
    for (int t0 = 0; t0 < T; t0 += 1024) {
        const int token = t0 + tid;
        const bool pred =
            (token < T) && (dispatch[(size_t)token * E_DIM + e] > 0.0f);

        const unsigned int mask = __builtin_amdgcn_ballot_w32(pred);
        if (lane == 0) waveCnt[wave] = __popc(mask);
        __syncthreads();

        int waveOff = 0, chunkTotal = 0;
        #pragma unroll
        for (int w = 0; w < 32; ++w) {
            const int c = waveCnt[w];
            if (w < wave) waveOff += c;
            chunkTotal += c;
        }
        if (pred) {
            const int pos =
                base + waveOff + __popc(mask & ((1u << lane) - 1u));
            out_idx[(size_t)e * T + pos] = token;
        }
        base += chunkTotal;
        __syncthreads();
    }

    // Pad the tail of this expert's row with -1.
    for (int i = base + tid; i < T; i += 1024)
        out_idx[(size_t)e * T + i] = -1;
}

extern "C" void kernel_launch(void* const* d_in, const int* in_sizes, int n_in,
                              void* d_out, int out_size, void* d_ws, size_t ws_size,
                              hipStream_t stream)
{
    const float* x = (const float*)d_in[0];   // (T, D) fp32
    const float* W = (const float*)d_in[1];   // (E, D) fp32
    const float* b = (const float*)d_in[2];   // (E,)   fp32

    const int T = in_sizes[0] / D_DIM;        // 16384

    float* logits   = (float*)d_out;                       // T*E
    float* dispatch = logits + (size_t)T * E_DIM;          // T*E
    int*   eidx     = (int*)(dispatch + (size_t)T * E_DIM);// E*T (int32)

    router_gemm_top2<<<dim3(T / TOK_PER_BLOCK), dim3(THREADS), 0, stream>>>(
        x, W, b, logits, dispatch, T);

    expert_index_scan<<<dim3(E_DIM), dim3(1024), 0, stream>>>(
        dispatch, eidx, T);
}